// EnsembleModel_86766929314358
// MI455X (gfx1250) — compile-verified
//
#include <hip/hip_runtime.h>
#include <stdint.h>

typedef float f4 __attribute__((ext_vector_type(4)));

#define TPB   1024
#define HW    262144              // 512*512 floats per (b,c) slice
#define CHUNK 65536               // floats per reduction block (4 blocks/slice)
#define NIT   (CHUNK / (TPB * 4)) // 16 iterations of 1024 x float4

// ---- CDNA5 async global->LDS copy (16B per lane), GVS addressing ----
__device__ __forceinline__ void async_ld16(uint32_t lds_addr, const float* sbase,
                                           uint32_t voff) {
  asm volatile("global_load_async_to_lds_b128 %0, %1, %2"
               :: "v"(lds_addr), "v"(voff), "s"(sbase)
               : "memory");
}
__device__ __forceinline__ void wait_async_le1() {
  asm volatile("s_wait_asynccnt 0x1" ::: "memory");
}
__device__ __forceinline__ void wait_async_0() {
  asm volatile("s_wait_asynccnt 0x0" ::: "memory");
}
__device__ __forceinline__ void wait_ds_0() {
  asm volatile("s_wait_dscnt 0x0" ::: "memory");
}

// Kernel 1: per-chunk partial sums of images (4 chunks per slice).
// Async-copies 16KB tiles into a double-buffered LDS stage, reads back,
// accumulates, then wave32 shuffle-reduces.
__global__ __launch_bounds__(TPB) void slice_partial_sums(
    const float* __restrict__ images, float* __restrict__ partials) {
  __shared__ f4 stage[2][TPB];
  __shared__ float wsum[TPB / 32];

  const uint32_t tid = threadIdx.x;
  const float* base = images + (size_t)blockIdx.x * CHUNK;
  const uint32_t lds0 = (uint32_t)(uintptr_t)&stage[0][tid];
  const uint32_t lds1 = (uint32_t)(uintptr_t)&stage[1][tid];

  async_ld16(lds0, base, tid * 16u);             // preload iter 0 -> buf 0

  f4 acc = {0.0f, 0.0f, 0.0f, 0.0f};
  for (int it = 0; it < NIT; ++it) {
    if (it + 1 < NIT) {
      // target buffer was last READ at iter it-1: make sure that ds_load
      // retired before the async engine may overwrite it (WAR hazard;
      // async LDS writes are unordered vs DS ops).
      wait_ds_0();
      async_ld16(((it + 1) & 1) ? lds1 : lds0, base,
                 (uint32_t)(it + 1) * (TPB * 16u) + tid * 16u);
      wait_async_le1();                          // current tile landed
    } else {
      wait_async_0();
    }
    f4 v = stage[it & 1][tid];
    acc += v;
  }

  float s = (acc.x + acc.y) + (acc.z + acc.w);
  #pragma unroll
  for (int off = 16; off > 0; off >>= 1) s += __shfl_down(s, off, 32);
  if ((tid & 31u) == 0u) wsum[tid >> 5] = s;
  __syncthreads();
  if (tid < 32u) {
    float t = wsum[tid];
    #pragma unroll
    for (int off = 16; off > 0; off >>= 1) t += __shfl_down(t, off, 32);
    if (tid == 0u) partials[blockIdx.x] = t;
  }
}

// Kernel 2: combine 4 partials per slice into the slice mean (deterministic).
__global__ void combine_means(const float* __restrict__ partials,
                              float* __restrict__ means) {
  const int s = threadIdx.x;
  const float m = ((partials[4 * s + 0] + partials[4 * s + 1]) +
                   (partials[4 * s + 2] + partials[4 * s + 3])) *
                  (1.0f / (float)HW);
  means[s] = m;
}

// Kernel 3: fused mask + shot-noise + quantize. One float4 per thread.
// 256 threads/block, 256 blocks per slice -> slice index is block-uniform.
__global__ __launch_bounds__(256) void shot_noise_quantize(
    const float* __restrict__ images, const float* __restrict__ noise1,
    const float* __restrict__ noise2, const float* __restrict__ means,
    float* __restrict__ out) {
  const uint32_t tid = threadIdx.x;
  const uint32_t gb  = blockIdx.x;
  const size_t   g   = (size_t)gb * 256 + tid;     // float4 index
  const uint32_t slice = gb >> 8;                  // 65536/256 blocks per slice
  const float inv = 1.0f / means[slice];           // block-uniform -> s_load

  const uint32_t l  = (uint32_t)((g << 2) & (HW - 1)); // float idx in slice
  const uint32_t i  = l >> 9;                          // row -> X
  const uint32_t j0 = l & 511u;                        // col -> Y

  const float step = 2.0f / 511.0f;                    // matches f32 linspace
  const float X  = __fadd_rn(-1.0f, __fmul_rn((float)i, step));
  const float aX = fabsf(X);
  const float X2 = __fmul_rn(X, X);

  const f4 img = ((const f4*)images)[g];
  const f4 v1  = __builtin_nontemporal_load((const f4*)noise1 + g);
  const f4 v2  = __builtin_nontemporal_load((const f4*)noise2 + g);

  f4 r;
  #pragma unroll
  for (int k = 0; k < 4; ++k) {
    const float Y  = __fadd_rn(-1.0f, __fmul_rn((float)(j0 + (uint32_t)k), step));
    const float R  = sqrtf(__fadd_rn(X2, __fmul_rn(Y, Y)));
    const bool  m  = (R > 0.95f) | (R < 0.19f) | (aX < 0.025f) | (fabsf(Y) < 0.025f);
    const float Bv = m ? 0.0f : img[k] * inv;
    const float ab = 10000.0f * Bv;                  // AMP * B
    const float A  = ((100.0f + 100.0f * v1[k]) + ab) + sqrtf(ab) * v2[k];
    float C = rintf((A / 200000.0f) * 65536.0f);     // round-half-even
    C = (A > 200000.0f) ? 200000.0f : C;
    C = (A < 0.0f) ? 0.0f : C;
    r[k] = C;
  }
  __builtin_nontemporal_store(r, (f4*)out + g);
}

extern "C" void kernel_launch(void* const* d_in, const int* in_sizes, int n_in,
                              void* d_out, int out_size, void* d_ws, size_t ws_size,
                              hipStream_t stream) {
  const float* images = (const float*)d_in[0];
  const float* noise1 = (const float*)d_in[1];
  const float* noise2 = (const float*)d_in[2];
  float* out = (float*)d_out;

  const int n       = in_sizes[0];   // 33554432
  const int nslices = n / HW;        // 128
  const int nchunks = n / CHUNK;     // 512 (4 per slice)

  float* partials = (float*)d_ws;          // nchunks floats
  float* means    = partials + nchunks;    // nslices floats

  slice_partial_sums<<<nchunks, TPB, 0, stream>>>(images, partials);
  combine_means<<<1, nslices, 0, stream>>>(partials, means);

  const int blocksB = (n / 4) / 256;       // 32768 blocks, one f4 per thread
  shot_noise_quantize<<<blocksB, 256, 0, stream>>>(images, noise1, noise2,
                                                   means, out);
}